// Encoder_43069932044748
// MI455X (gfx1250) — compile-verified
//
#include <hip/hip_runtime.h>
#include <hip/hip_bf16.h>

typedef float v2f __attribute__((ext_vector_type(2)));
typedef float v8f __attribute__((ext_vector_type(8)));

static constexpr int NN = 100000;   // nodes
static constexpr int NE = 1600000;  // edges
static constexpr int D  = 128;      // feature dim (both in and hidden)

// ---------------------------------------------------------------- utilities
__global__ void zero_kernel(float4* __restrict__ p, long n4) {
    long i = (long)blockIdx.x * blockDim.x + threadIdx.x;
    long stride = (long)gridDim.x * blockDim.x;
    float4 z = {0.f, 0.f, 0.f, 0.f};
    for (; i < n4; i += stride) p[i] = z;
}

// deg[dst] += ew   (self-loop +1 added later)
__global__ void deg_scatter_kernel(const int* __restrict__ dst,
                                   const float* __restrict__ ew,
                                   float* __restrict__ deg, int E) {
    int e = blockIdx.x * blockDim.x + threadIdx.x;
    if (e < E) atomicAdd(&deg[dst[e]], ew[e]);
}

// deg -> deg^{-1/2} in place (deg+1 >= 1 always, so no zero guard needed)
__global__ void dis_kernel(float* __restrict__ deg, int n) {
    int i = blockIdx.x * blockDim.x + threadIdx.x;
    if (i < n) deg[i] = rsqrtf(deg[i] + 1.0f);
}

// norm[e] = dis[src] * ew * dis[dst]  (graph-constant, reused by all layers)
__global__ void norm_kernel(const int* __restrict__ src, const int* __restrict__ dst,
                            const float* __restrict__ ew, const float* __restrict__ dis,
                            float* __restrict__ norm, int E) {
    int e = blockIdx.x * blockDim.x + threadIdx.x;
    if (e < E) norm[e] = dis[src[e]] * ew[e] * dis[dst[e]];
}

// ---------------------------------------------------------------- GEMM (WMMA f32)
// H[M x 128] = X[M x 128] * W[128 x 128], row-major.
// Block = 256 threads = 8 waves. Block owns a 16-row stripe; wave w owns
// columns [16w, 16w+16). K swept in 32 steps of V_WMMA_F32_16X16X4_F32.
//
// Fragment layouts per CDNA5 ISA 7.12.2:
//   A (16x4 f32, 2 VGPRs): v0 = K={0 | 2} for lane halves, v1 = K={1 | 3}
//   B (4x16 f32, 2 VGPRs): row K striped across lanes, same half split
//   C/D (16x16 f32, 8 VGPRs): VGPR j = row M=j (lanes 0-15), M=j+8 (lanes 16-31)
__global__ __launch_bounds__(256)
void gemm_wmma_f32(const float* __restrict__ X, const float* __restrict__ W,
                   float* __restrict__ H) {
    const int wave = threadIdx.x >> 5;
    const int lane = threadIdx.x & 31;
    const int half = lane >> 4;     // 0: lanes 0-15, 1: lanes 16-31
    const int r    = lane & 15;
    const int m0   = blockIdx.x * 16;
    const int n0   = wave * 16;

    const float* __restrict__ xrow = X + (size_t)(m0 + r) * D;

    v8f acc = {};
#pragma unroll
    for (int k0 = 0; k0 < D; k0 += 4) {
        const int ka = k0 + 2 * half;
        v2f a, b;
        a.x = xrow[ka];
        a.y = xrow[ka + 1];
        b.x = W[(size_t)ka * D + n0 + r];
        b.y = W[(size_t)(ka + 1) * D + n0 + r];
        acc = __builtin_amdgcn_wmma_f32_16x16x4_f32(
            /*neg_a=*/false, a, /*neg_b=*/false, b,
            /*c_mod=*/(short)0, acc, /*reuse_a=*/false, /*reuse_b=*/false);
    }

#pragma unroll
    for (int j = 0; j < 8; ++j)
        H[(size_t)(m0 + j + 8 * half) * D + n0 + r] = acc[j];
}

// ---------------------------------------------------------------- edge scatter
// One wave per edge; each lane handles 4 consecutive features (float4 gather of
// the L2-resident h[src] row, then 4 f32 atomics into agg[dst]).
__global__ __launch_bounds__(256)
void scatter_kernel(const int* __restrict__ src, const int* __restrict__ dst,
                    const float* __restrict__ norm, const float* __restrict__ H,
                    float* __restrict__ AGG, int E) {
    long gid = (long)blockIdx.x * blockDim.x + threadIdx.x;
    int e    = (int)(gid >> 5);
    int lane = (int)(gid & 31);
    if (e >= E) return;
    int   s = src[e];
    int   d = dst[e];
    float w = norm[e];
    float4 v = ((const float4*)(H + (size_t)s * D))[lane];
    float* out = AGG + (size_t)d * D + lane * 4;
    atomicAdd(out + 0, w * v.x);
    atomicAdd(out + 1, w * v.y);
    atomicAdd(out + 2, w * v.z);
    atomicAdd(out + 3, w * v.w);
}

// ---------------------------------------------------------------- finalize
// out = prelu(agg + dis^2 * h + b, a), vectorized float4; one wave per node.
__global__ __launch_bounds__(256)
void finalize_kernel(const float* __restrict__ AGG, const float* __restrict__ H,
                     const float* __restrict__ dis, const float* __restrict__ b,
                     const float* __restrict__ a, float* __restrict__ OUT, int n) {
    long gid = (long)blockIdx.x * blockDim.x + threadIdx.x;
    int node = (int)(gid >> 5);
    int lane = (int)(gid & 31);
    if (node >= n) return;
    float di = dis[node];
    float sl = di * di;
    float4 av = ((const float4*)(AGG + (size_t)node * D))[lane];
    float4 hv = ((const float4*)(H + (size_t)node * D))[lane];
    float4 bv = ((const float4*)b)[lane];
    float4 al = ((const float4*)a)[lane];
    float4 t;
    t.x = av.x + sl * hv.x + bv.x;
    t.y = av.y + sl * hv.y + bv.y;
    t.z = av.z + sl * hv.z + bv.z;
    t.w = av.w + sl * hv.w + bv.w;
    t.x = t.x >= 0.f ? t.x : al.x * t.x;
    t.y = t.y >= 0.f ? t.y : al.y * t.y;
    t.z = t.z >= 0.f ? t.z : al.z * t.z;
    t.w = t.w >= 0.f ? t.w : al.w * t.w;
    ((float4*)(OUT + (size_t)node * D))[lane] = t;
}

// ---------------------------------------------------------------- launcher
extern "C" void kernel_launch(void* const* d_in, const int* in_sizes, int n_in,
                              void* d_out, int out_size, void* d_ws, size_t ws_size,
                              hipStream_t stream) {
    const float* x  = (const float*)d_in[0];
    const int*   ei = (const int*)d_in[1];   // [2, E] int32 (JAX default)
    const float* ew = (const float*)d_in[2];
    const float* Wm[3] = {(const float*)d_in[3], (const float*)d_in[6], (const float*)d_in[9]};
    const float* bm[3] = {(const float*)d_in[4], (const float*)d_in[7], (const float*)d_in[10]};
    const float* am[3] = {(const float*)d_in[5], (const float*)d_in[8], (const float*)d_in[11]};

    const int* src = ei;
    const int* dst = ei + NE;

    // workspace layout (floats): dis | norm | H | ping | pong   (~160 MB)
    float* dis  = (float*)d_ws;
    float* norm = dis + 102400;                 // N padded for alignment
    float* H    = norm + NE;
    float* A    = H + (size_t)NN * D;
    float* Bf   = A + (size_t)NN * D;
    float* out  = (float*)d_out;

    // ---- graph-constant normalization (computed once)
    zero_kernel<<<256, 256, 0, stream>>>((float4*)dis, 102400 / 4);
    deg_scatter_kernel<<<(NE + 255) / 256, 256, 0, stream>>>(dst, ew, dis, NE);
    dis_kernel<<<(NN + 255) / 256, 256, 0, stream>>>(dis, NN);
    norm_kernel<<<(NE + 255) / 256, 256, 0, stream>>>(src, dst, ew, dis, norm, NE);

    // ---- three GCN layers; last one aggregates straight into d_out
    const float* Xin = x;
    float* aggbuf[3] = {A, Bf, out};
    for (int l = 0; l < 3; ++l) {
        float* AGG = aggbuf[l];
        gemm_wmma_f32<<<NN / 16, 256, 0, stream>>>(Xin, Wm[l], H);
        zero_kernel<<<2048, 256, 0, stream>>>((float4*)AGG, (long)NN * D / 4);
        scatter_kernel<<<((long)NE * 32) / 256, 256, 0, stream>>>(src, dst, norm, H, AGG, NE);
        finalize_kernel<<<((long)NN * 32) / 256, 256, 0, stream>>>(AGG, H, dis, bm[l], am[l], AGG, NN);
        Xin = AGG;
    }
}